// SGCNet_41850161332536
// MI455X (gfx1250) — compile-verified
//
#include <hip/hip_runtime.h>
#include <hip/hip_bf16.h>

typedef __attribute__((ext_vector_type(2))) float v2f;
typedef __attribute__((ext_vector_type(8))) float v8f;

#define N_NODES 50000
#define IN_CH   128
#define OUT_CH  64
#define N_EDGES 800000

// -------- degree / normalization --------
__global__ void k_init_deg(float* __restrict__ deg) {
    int i = blockIdx.x * blockDim.x + threadIdx.x;
    if (i < N_NODES) deg[i] = 1.0f;                 // self-loop
}

__global__ void k_count_deg(const int* __restrict__ ei, float* __restrict__ deg) {
    int e = blockIdx.x * blockDim.x + threadIdx.x;
    if (e < N_EDGES) atomicAdd(&deg[ei[N_EDGES + e]], 1.0f);   // col = ei[1][e]
}

__global__ void k_rsqrt(float* __restrict__ dinv) {
    int i = blockIdx.x * blockDim.x + threadIdx.x;
    if (i < N_NODES) dinv[i] = rsqrtf(dinv[i]);     // deg >= 1 always
}

// -------- h0 = x @ W  via V_WMMA_F32_16X16X4_F32 (full fp32) --------
// 3125 M-tiles x 4 N-tiles = 12500 16x16 output tiles; one wave per tile.
__global__ void k_gemm_wmma(const float* __restrict__ x,
                            const float* __restrict__ W,
                            float* __restrict__ h0) {
    int wave = (int)((blockIdx.x * blockDim.x + threadIdx.x) >> 5);
    int lane = threadIdx.x & 31;
    if (wave >= 3125 * 4) return;                   // whole-wave exit: EXEC stays all-1s
    const int tm = wave >> 2;                       // 0..3124
    const int tn = wave & 3;                        // 0..3
    const int m    = lane & 15;
    const int hi   = lane >> 4;                     // 0 for lanes 0-15, 1 for 16-31
    const int koff = hi << 1;                       // K sub-offset {0,2}
    const int row  = tm * 16 + m;                   // A row (M = lane%16)
    const int ncol = tn * 16 + m;                   // B col (N = lane%16)

    v8f c = {};
    #pragma unroll 4
    for (int k0 = 0; k0 < IN_CH; k0 += 4) {
        v2f a, bf;
        const float* ap = x + (size_t)row * IN_CH + k0 + koff;
        a.x = ap[0];                                // A[M][k0+koff]
        a.y = ap[1];                                // A[M][k0+koff+1]
        bf.x = W[(size_t)(k0 + koff) * OUT_CH + ncol];      // B[k0+koff][N]
        bf.y = W[(size_t)(k0 + koff + 1) * OUT_CH + ncol];  // B[k0+koff+1][N]
        // 8 args: (neg_a, A, neg_b, B, c_mod, C, reuse_a, reuse_b)
        c = __builtin_amdgcn_wmma_f32_16x16x4_f32(false, a, false, bf,
                                                  (short)0, c, false, false);
    }
    // C/D layout: VGPR j -> M = j + 8*hi, N = lane%16
    const int mbase = tm * 16 + (hi << 3);
    const int n = tn * 16 + (lane & 15);
    #pragma unroll
    for (int j = 0; j < 8; ++j)
        h0[(size_t)(mbase + j) * OUT_CH + n] = c[j];
}

// -------- per-node init: hout = scale_node * hin  (self-loop term) --------
__global__ void k_self(const float* __restrict__ hin, const float* __restrict__ dinv,
                       float* __restrict__ hout) {
    int idx = blockIdx.x * blockDim.x + threadIdx.x;    // over N_NODES * 32 float2's
    if (idx >= N_NODES * (OUT_CH / 2)) return;
    int node = idx >> 5;
    float d = dinv[node];
    float s = d * d;
    const float2* vi = (const float2*)hin;
    float2* vo = (float2*)hout;
    float2 v = vi[idx];
    vo[idx] = make_float2(s * v.x, s * v.y);
}

// -------- final init: out = b + scale_node * h1 --------
__global__ void k_out_init(const float* __restrict__ h1, const float* __restrict__ dinv,
                           const float* __restrict__ b, float* __restrict__ out) {
    int idx = blockIdx.x * blockDim.x + threadIdx.x;
    if (idx >= N_NODES * (OUT_CH / 2)) return;
    int node = idx >> 5;
    int pair = idx & 31;
    float d = dinv[node];
    float s = d * d;
    const float2* vi = (const float2*)h1;
    const float2* bv = (const float2*)b;
    float2* vo = (float2*)out;
    float2 v = vi[idx];
    float2 bb = bv[pair];
    vo[idx] = make_float2(bb.x + s * v.x, bb.y + s * v.y);
}

// -------- edge scatter: one wave per edge, float2 of channels per lane --------
__global__ void k_prop(const float* __restrict__ hin, float* __restrict__ hout,
                       const int* __restrict__ ei, const float* __restrict__ dinv) {
    int wave = (int)((blockIdx.x * blockDim.x + threadIdx.x) >> 5);
    int lane = threadIdx.x & 31;
    if (wave >= N_EDGES) return;
    int r = ei[wave];                 // source (row)
    int c = ei[N_EDGES + wave];       // target (col)
    float norm = dinv[r] * dinv[c];
    const float2* src = (const float2*)(hin + (size_t)r * OUT_CH);
    float2 v = src[lane];             // 256B coalesced gather per wave (L2-resident)
    float* dst = hout + (size_t)c * OUT_CH + 2 * lane;
    atomicAdd(dst,     norm * v.x);
    atomicAdd(dst + 1, norm * v.y);
}

extern "C" void kernel_launch(void* const* d_in, const int* in_sizes, int n_in,
                              void* d_out, int out_size, void* d_ws, size_t ws_size,
                              hipStream_t stream) {
    const float* x  = (const float*)d_in[0];
    const int*   ei = (const int*)d_in[1];      // [2, N_EDGES] int32, row-major
    const float* W  = (const float*)d_in[2];    // [128, 64]
    const float* b  = (const float*)d_in[3];    // [64]
    float* out = (float*)d_out;                 // [50000, 64]

    float* ws   = (float*)d_ws;
    float* h0   = ws;                                    // 3.2M floats
    float* h1   = ws + (size_t)N_NODES * OUT_CH;         // 3.2M floats
    float* dinv = h1 + (size_t)N_NODES * OUT_CH;         // 50k floats (deg -> dinv in place)

    const int B = 256;
    // 1) degree (self-loops included) -> dinv
    k_init_deg <<<(N_NODES + B - 1) / B, B, 0, stream>>>(dinv);
    k_count_deg<<<(N_EDGES + B - 1) / B, B, 0, stream>>>(ei, dinv);
    k_rsqrt    <<<(N_NODES + B - 1) / B, B, 0, stream>>>(dinv);

    // 2) h0 = x @ W  (WMMA fp32; 12500 tiles, 8 waves/block)
    k_gemm_wmma<<<(12500 + 7) / 8, B, 0, stream>>>(x, W, h0);

    // 3) hop 1: h1 = self-term, then scatter edge messages h0 -> h1
    const int nvec = N_NODES * (OUT_CH / 2);
    k_self<<<(nvec + B - 1) / B, B, 0, stream>>>(h0, dinv, h1);
    k_prop<<<(N_EDGES * 32 + B - 1) / B, B, 0, stream>>>(h0, h1, ei, dinv);

    // 4) hop 2: out = b + self-term, then scatter edge messages h1 -> out
    k_out_init<<<(nvec + B - 1) / B, B, 0, stream>>>(h1, dinv, b, out);
    k_prop<<<(N_EDGES * 32 + B - 1) / B, B, 0, stream>>>(h1, out, ei, dinv);
}